// LightGCN_54400055771920
// MI455X (gfx1250) — compile-verified
//
#include <hip/hip_runtime.h>

#define DIM 128
#define TILE 8               // edges gathered per wave per stage
#define WAVES_PER_BLOCK 4
#define SPMM_BLOCK (WAVES_PER_BLOCK * 32)
#define SPMM_GRID 2048
#define EW_BLOCK 256
#define EW_GRID 4096

typedef float4 f4;

// Low 32 bits of a generic (flat) pointer into LDS == wave-relative LDS byte
// offset (addrspacecast local->flat keeps the AS(3) offset in addr[31:0]).
__device__ __forceinline__ unsigned lds_off(const void* p) {
    return (unsigned)(unsigned long long)p;
}

__global__ __launch_bounds__(SPMM_BLOCK)
void spmm_kernel(const int* __restrict__ src, const int* __restrict__ dst,
                 const float* __restrict__ vals, const float* __restrict__ h,
                 float* __restrict__ hn, int nnz) {
    // [stage][wave][edge][lane] : 2*4*8*32*16B = 32 KB
    __shared__ f4 gbuf[2][WAVES_PER_BLOCK][TILE][32];

    const int wave_in_blk = threadIdx.x >> 5;
    const int lane        = threadIdx.x & 31;
    const int gwave       = blockIdx.x * WAVES_PER_BLOCK + wave_in_blk;
    const int total_waves = gridDim.x * WAVES_PER_BLOCK;

    const int nchunks = (nnz + TILE - 1) / TILE;
    const int cpw     = (nchunks + total_waves - 1) / total_waves;
    const int c_begin = gwave * cpw;
    const int c_end   = min(c_begin + cpw, nchunks);
    if (c_begin >= c_end) return;

    unsigned lbase[2];
    lbase[0] = lds_off(&gbuf[0][wave_in_blk][0][lane]);
    lbase[1] = lds_off(&gbuf[1][wave_in_blk][0][lane]);

    // Issue TILE per-lane async row gathers: one global_load_async_to_lds_b128
    // per edge moves a whole 512B row (32 lanes x float4) into LDS.
    auto issue = [&](int c, unsigned base) {
        const int e0 = c * TILE;
#pragma unroll
        for (int i = 0; i < TILE; ++i) {
            int e = e0 + i;
            e = e < nnz ? e : nnz - 1;          // clamp: keep op count uniform
            const int d = dst[e];
            const float* g = h + (size_t)d * DIM + (lane << 2);
            const unsigned l = base + (unsigned)(i * 32 * sizeof(f4));
            asm volatile("global_load_async_to_lds_b128 %0, %1, off"
                         :: "v"(l), "v"(g) : "memory");
        }
    };

    auto process = [&](int c, int stage) {
        const int e0 = c * TILE;
        const f4* buf = &gbuf[stage][wave_in_blk][0][lane];
#pragma unroll
        for (int i = 0; i < TILE; ++i) {
            const int e = e0 + i;
            if (e >= nnz) break;                 // wave-uniform
            const float v = vals[e];
            const int   s = src[e];
            const f4 x = buf[i * 32];
            float* o = hn + (size_t)s * DIM + (lane << 2);
            unsafeAtomicAdd(o + 0, x.x * v);
            unsafeAtomicAdd(o + 1, x.y * v);
            unsafeAtomicAdd(o + 2, x.z * v);
            unsafeAtomicAdd(o + 3, x.w * v);
        }
    };

    int stage = 0;
    issue(c_begin, lbase[0]);
    for (int c = c_begin; c < c_end; ++c) {
        const bool has_next = (c + 1) < c_end;
        if (has_next) {
            issue(c + 1, lbase[stage ^ 1]);
            // 16 outstanding, in-order completion: <=8 left => current done
            asm volatile("s_wait_asynccnt 8" ::: "memory");
        } else {
            asm volatile("s_wait_asynccnt 0" ::: "memory");
        }
        process(c, stage);
        stage ^= 1;
    }
}

__global__ __launch_bounds__(EW_BLOCK)
void init_kernel(const float* __restrict__ ue, const float* __restrict__ ie,
                 float* __restrict__ acc, float* __restrict__ h,
                 int nuser_f, int ntot_f) {
    const int n4  = ntot_f >> 2;
    const int nu4 = nuser_f >> 2;
    for (int i = blockIdx.x * blockDim.x + threadIdx.x; i < n4;
         i += gridDim.x * blockDim.x) {
        const f4 v = (i < nu4) ? ((const f4*)ue)[i] : ((const f4*)ie)[i - nu4];
        ((f4*)acc)[i] = v;
        ((f4*)h)[i]   = v;
    }
}

__global__ __launch_bounds__(EW_BLOCK)
void zero_kernel(float* __restrict__ p, int n4) {
    const f4 z = {0.f, 0.f, 0.f, 0.f};
    for (int i = blockIdx.x * blockDim.x + threadIdx.x; i < n4;
         i += gridDim.x * blockDim.x)
        ((f4*)p)[i] = z;
}

__global__ __launch_bounds__(EW_BLOCK)
void addscale_kernel(float* __restrict__ acc, const float* __restrict__ hn,
                     int n4, float scale) {
    for (int i = blockIdx.x * blockDim.x + threadIdx.x; i < n4;
         i += gridDim.x * blockDim.x) {
        f4 a = ((f4*)acc)[i];
        const f4 b = ((const f4*)hn)[i];
        a.x = (a.x + b.x) * scale;
        a.y = (a.y + b.y) * scale;
        a.z = (a.z + b.z) * scale;
        a.w = (a.w + b.w) * scale;
        ((f4*)acc)[i] = a;
    }
}

extern "C" void kernel_launch(void* const* d_in, const int* in_sizes, int n_in,
                              void* d_out, int out_size, void* d_ws, size_t ws_size,
                              hipStream_t stream) {
    const int*   adj_src = (const int*)d_in[0];
    const int*   adj_dst = (const int*)d_in[1];
    const float* adj_val = (const float*)d_in[2];
    const float* ue      = (const float*)d_in[3];
    const float* ie      = (const float*)d_in[4];

    const int nnz     = in_sizes[0];
    const int nuser_f = in_sizes[3];                 // NUM_USERS*DIM floats
    const int ntot_f  = in_sizes[3] + in_sizes[4];   // N*DIM floats
    const int n4      = ntot_f >> 2;

    float* acc = (float*)d_out;
    float* h   = (float*)d_ws;                       // ping
    float* hn  = (float*)d_ws + (size_t)ntot_f;      // pong

    const int ew_grid = min((n4 + EW_BLOCK - 1) / EW_BLOCK, EW_GRID);

    init_kernel<<<ew_grid, EW_BLOCK, 0, stream>>>(ue, ie, acc, h, nuser_f, ntot_f);

    for (int layer = 0; layer < 3; ++layer) {
        zero_kernel<<<ew_grid, EW_BLOCK, 0, stream>>>(hn, n4);
        spmm_kernel<<<SPMM_GRID, SPMM_BLOCK, 0, stream>>>(adj_src, adj_dst,
                                                          adj_val, h, hn, nnz);
        const float scale = (layer == 2) ? 0.25f : 1.0f;
        addscale_kernel<<<ew_grid, EW_BLOCK, 0, stream>>>(acc, hn, n4, scale);
        float* t = h; h = hn; hn = t;
    }
}